// gatv2block_70600672411871
// MI455X (gfx1250) — compile-verified
//
#include <hip/hip_runtime.h>
#include <hip/hip_bf16.h>

#define N_NODES  100000
#define E_EDGES  1600000
#define IN_F     256
#define HD       128          // H*D
#define H_HEADS  8
#define SLOPE    0.2f

#define ROWS_PB    64
#define LDS_STRIDE 257        // 257 % 64 banks == 1 -> conflict-free A reads

typedef __attribute__((ext_vector_type(2))) float v2f;
typedef __attribute__((ext_vector_type(8))) float v8f;

__device__ __forceinline__ unsigned f2u(float f) {
    unsigned u = __float_as_uint(f);
    return (u & 0x80000000u) ? ~u : (u | 0x80000000u);
}
__device__ __forceinline__ float u2f(unsigned u) {
    u = (u & 0x80000000u) ? (u & 0x7FFFFFFFu) : ~u;
    return __uint_as_float(u);
}

// ---------------------------------------------------------------------------
// Kernel 1: fused triple GEMM  el = xWs+bs, er = xWd+bd, res = xWr+br
// 64 rows per block, 8 waves; wave w owns cols [16w,16w+16) of all 3 outputs.
// fp32 WMMA: V_WMMA_F32_16X16X4_F32, K-loop of 64 steps.
// ---------------------------------------------------------------------------
__global__ __launch_bounds__(256) void gat_gemm3(
    const float* __restrict__ x,
    const float* __restrict__ Wsrc, const float* __restrict__ bsrc,
    const float* __restrict__ Wdst, const float* __restrict__ bdst,
    const float* __restrict__ Wres, const float* __restrict__ bres,
    float* __restrict__ el, float* __restrict__ er, float* __restrict__ outres)
{
    __shared__ float xs[ROWS_PB * LDS_STRIDE];

    const int tid  = threadIdx.x;
    const int row0 = blockIdx.x * ROWS_PB;

    // Stage 64x256 fp32 tile of x into LDS (zero-pad tail rows).
    for (int i = tid; i < ROWS_PB * (IN_F / 4); i += 256) {
        const int r  = i >> 6;       // i / (IN_F/4)
        const int c4 = i & 63;
        float4 v = make_float4(0.f, 0.f, 0.f, 0.f);
        const int gr = row0 + r;
        if (gr < N_NODES)
            v = *(const float4*)(x + (size_t)gr * IN_F + c4 * 4);
        float* p = &xs[r * LDS_STRIDE + c4 * 4];
        p[0] = v.x; p[1] = v.y; p[2] = v.z; p[3] = v.w;
    }
    __syncthreads();

    const int wave    = tid >> 5;
    const int lane    = tid & 31;
    const int l15     = lane & 15;
    const int khalf   = (lane >> 4) * 2;   // lanes 0-15: K+0/1, lanes 16-31: K+2/3
    const int rhalf   = (lane >> 4) * 8;   // D-matrix: lanes 16-31 hold rows M+8..M+15
    const int colbase = wave * 16;

    const float* const Wm[3] = { Wsrc, Wdst, Wres };
    const float* const bm[3] = { bsrc, bdst, bres };
    float* const       Om[3] = { el,   er,   outres };

#pragma unroll
    for (int m = 0; m < 3; ++m) {
        const float* __restrict__ W = Wm[m];
        v8f acc0 = {}; v8f acc1 = {}; v8f acc2 = {}; v8f acc3 = {};

        for (int k = 0; k < IN_F; k += 4) {
            // B fragment (4x16, K-major halves across lane groups)
            v2f b;
            b.x = W[(k + khalf)     * HD + colbase + l15];
            b.y = W[(k + khalf + 1) * HD + colbase + l15];

#define GAT_WSTEP(ACC, RT)                                                     \
            {                                                                  \
                v2f a;                                                         \
                const int ab = ((RT) * 16 + l15) * LDS_STRIDE + k + khalf;     \
                a.x = xs[ab];                                                  \
                a.y = xs[ab + 1];                                              \
                ACC = __builtin_amdgcn_wmma_f32_16x16x4_f32(                   \
                        false, a, false, b, (short)0, ACC, false, false);      \
            }
            GAT_WSTEP(acc0, 0)
            GAT_WSTEP(acc1, 1)
            GAT_WSTEP(acc2, 2)
            GAT_WSTEP(acc3, 3)
#undef GAT_WSTEP
        }

        const float bias = bm[m][colbase + l15];
        float* __restrict__ O = Om[m];

#define GAT_STORE(ACC, RT)                                                     \
        {                                                                      \
            _Pragma("unroll")                                                  \
            for (int j = 0; j < 8; ++j) {                                      \
                const int r = row0 + (RT) * 16 + rhalf + j;                    \
                if (r < N_NODES)                                               \
                    O[(size_t)r * HD + colbase + l15] = ACC[j] + bias;         \
            }                                                                  \
        }
        GAT_STORE(acc0, 0)
        GAT_STORE(acc1, 1)
        GAT_STORE(acc2, 2)
        GAT_STORE(acc3, 3)
#undef GAT_STORE
    }
}

// ---------------------------------------------------------------------------
// Kernel 2: per-edge attention logits + segment max (encoded-uint atomicMax).
// One wave32 per edge; lane covers 4 channels; 4 lanes per head.
// ---------------------------------------------------------------------------
__global__ __launch_bounds__(256) void gat_score(
    const float* __restrict__ el, const float* __restrict__ er,
    const int* __restrict__ src, const int* __restrict__ dst,
    const float* __restrict__ attn,
    float* __restrict__ score, unsigned* __restrict__ maxu)
{
    const int e = blockIdx.x * 8 + (threadIdx.x >> 5);
    if (e >= E_EDGES) return;
    const int lane = threadIdx.x & 31;
    const int u = src[e];
    const int v = dst[e];
    const int c = lane * 4;

    const float4 a4 = *(const float4*)(el   + (size_t)u * HD + c);
    const float4 b4 = *(const float4*)(er   + (size_t)v * HD + c);
    const float4 w4 = *(const float4*)(attn + c);

    float p = 0.f, t;
    t = a4.x + b4.x; t = (t > 0.f) ? t : SLOPE * t; p += t * w4.x;
    t = a4.y + b4.y; t = (t > 0.f) ? t : SLOPE * t; p += t * w4.y;
    t = a4.z + b4.z; t = (t > 0.f) ? t : SLOPE * t; p += t * w4.z;
    t = a4.w + b4.w; t = (t > 0.f) ? t : SLOPE * t; p += t * w4.w;

    p += __shfl_xor(p, 1, 32);
    p += __shfl_xor(p, 2, 32);

    if ((lane & 3) == 0) {
        const int h = lane >> 2;
        score[(size_t)e * H_HEADS + h] = p;
        atomicMax(&maxu[(size_t)v * H_HEADS + h], f2u(p));
    }
}

// ---------------------------------------------------------------------------
// Kernel 3: a = exp(s - m[dst]) (in place) and denom[dst] += a (hw f32 atomic)
// ---------------------------------------------------------------------------
__global__ __launch_bounds__(256) void gat_expsum(
    const int* __restrict__ dst, float* __restrict__ score,
    const unsigned* __restrict__ maxu, float* __restrict__ denom)
{
    const long long t = (long long)blockIdx.x * 256 + threadIdx.x;
    const int e = (int)(t >> 3);
    const int h = (int)(t & 7);
    if (e >= E_EDGES) return;
    const int v = dst[e];
    const float m = u2f(maxu[(size_t)v * H_HEADS + h]);
    const float a = __expf(score[t] - m);
    score[t] = a;
    unsafeAtomicAdd(&denom[(size_t)v * H_HEADS + h], a);
}

// ---------------------------------------------------------------------------
// Kernel 4: out[v] += (a/denom[v]) * el[u]   (fp32 hardware atomics into L2)
// One wave32 per edge; lane covers 4 channels.
// ---------------------------------------------------------------------------
__global__ __launch_bounds__(256) void gat_aggregate(
    const float* __restrict__ el,
    const int* __restrict__ src, const int* __restrict__ dst,
    const float* __restrict__ alpha, const float* __restrict__ denom,
    float* __restrict__ out)
{
    const int e = blockIdx.x * 8 + (threadIdx.x >> 5);
    if (e >= E_EDGES) return;
    const int lane = threadIdx.x & 31;
    const int u = src[e];
    const int v = dst[e];
    const int h = lane >> 2;
    const int c = lane * 4;

    const float w  = alpha[(size_t)e * H_HEADS + h] / denom[(size_t)v * H_HEADS + h];
    const float4 f = *(const float4*)(el + (size_t)u * HD + c);

    float* o = out + (size_t)v * HD + c;
    unsafeAtomicAdd(o + 0, w * f.x);
    unsafeAtomicAdd(o + 1, w * f.y);
    unsafeAtomicAdd(o + 2, w * f.z);
    unsafeAtomicAdd(o + 3, w * f.w);
}

// ---------------------------------------------------------------------------
extern "C" void kernel_launch(void* const* d_in, const int* in_sizes, int n_in,
                              void* d_out, int out_size, void* d_ws, size_t ws_size,
                              hipStream_t stream) {
    const float* x    = (const float*)d_in[0];
    const int*   src  = (const int*)  d_in[1];
    const int*   dst  = (const int*)  d_in[2];
    const float* Wsrc = (const float*)d_in[3];
    const float* bsrc = (const float*)d_in[4];
    const float* Wdst = (const float*)d_in[5];
    const float* bdst = (const float*)d_in[6];
    const float* attn = (const float*)d_in[7];
    const float* Wres = (const float*)d_in[8];
    const float* bres = (const float*)d_in[9];
    float* out = (float*)d_out;

    // workspace layout
    char* ws = (char*)d_ws;
    const size_t elBytes    = (size_t)N_NODES * HD * sizeof(float);   // 51.2 MB
    const size_t scoreBytes = (size_t)E_EDGES * H_HEADS * sizeof(float);
    const size_t nhBytes    = (size_t)N_NODES * H_HEADS * sizeof(float);
    float*    el    = (float*)   (ws);
    float*    er    = (float*)   (ws + elBytes);
    float*    score = (float*)   (ws + 2 * elBytes);
    unsigned* maxu  = (unsigned*)(ws + 2 * elBytes + scoreBytes);
    float*    denom = (float*)   (ws + 2 * elBytes + scoreBytes + nhBytes);

    // zero segment-reduction buffers (graph-capture-safe)
    hipMemsetAsync(maxu,  0, nhBytes, stream);   // 0 encodes -inf ordering floor
    hipMemsetAsync(denom, 0, nhBytes, stream);

    // 1) fused WMMA GEMMs: el, er (-> ws), res (-> d_out accumulation base)
    const int gemmBlocks = (N_NODES + ROWS_PB - 1) / ROWS_PB;
    gat_gemm3<<<gemmBlocks, 256, 0, stream>>>(x, Wsrc, bsrc, Wdst, bdst,
                                              Wres, bres, el, er, out);

    // 2) edge logits + segment max
    const int edgeBlocks = (E_EDGES + 7) / 8;
    gat_score<<<edgeBlocks, 256, 0, stream>>>(el, er, src, dst, attn, score, maxu);

    // 3) exp + denom
    const long long ehThreads = (long long)E_EDGES * H_HEADS;
    const int expBlocks = (int)((ehThreads + 255) / 256);
    gat_expsum<<<expBlocks, 256, 0, stream>>>(dst, score, maxu, denom);

    // 4) weighted scatter-aggregate into out (already holds residual term)
    gat_aggregate<<<edgeBlocks, 256, 0, stream>>>(el, src, dst, score, denom, out);
}